// GCN2_48593259987025
// MI455X (gfx1250) — compile-verified
//
#include <hip/hip_runtime.h>

typedef float v2f __attribute__((ext_vector_type(2)));
typedef float v8f __attribute__((ext_vector_type(8)));

#define FDIM 128
#define HDIM 128
#define NGRAPH 128
#define NCLASS 10
#define NLAYER 8
#define LDS_STRIDE 132   // 128 + 4 pad: lane (lm,hi) -> DWORD banks 4*lm+2*hi+{0,1}, conflict-free

// Fused dual GEMM: out = relu?(h) @ Ws + bias ; m = relu?(h) @ Wr
// Block = 256 threads = 8 waves; block owns a 16-row strip, wave w owns cols [16w,16w+16).
// A strip (16x128) is staged once in LDS (relu fused into the fill).
template <bool RELU>
__global__ __launch_bounds__(256) void gemm_dual_wmma(
    const float* __restrict__ h, const float* __restrict__ Wr,
    const float* __restrict__ Ws, const float* __restrict__ bias,
    float* __restrict__ out, float* __restrict__ m, int N)
{
    __shared__ float As[16 * LDS_STRIDE];

    const int tid = threadIdx.x;
    const int r0  = blockIdx.x * 16;

    // ---- cooperative A-strip load: 16 rows x 128 floats, 8 floats/thread ----
    {
        const int row = tid >> 4;         // 0..15
        const int ch  = tid & 15;         // chunk of 8 floats
        int grow = r0 + row;
        if (grow >= N) grow = N - 1;
        const float4* src = (const float4*)(h + (size_t)grow * FDIM + ch * 8);
        float4 v0 = src[0];
        float4 v1 = src[1];
        if (RELU) {
            v0.x = fmaxf(v0.x, 0.0f); v0.y = fmaxf(v0.y, 0.0f);
            v0.z = fmaxf(v0.z, 0.0f); v0.w = fmaxf(v0.w, 0.0f);
            v1.x = fmaxf(v1.x, 0.0f); v1.y = fmaxf(v1.y, 0.0f);
            v1.z = fmaxf(v1.z, 0.0f); v1.w = fmaxf(v1.w, 0.0f);
        }
        float* d = As + row * LDS_STRIDE + ch * 8;
        *(float4*)(d)     = v0;
        *(float4*)(d + 4) = v1;
    }
    __syncthreads();

    const int wave = tid >> 5;            // 0..7 -> column tile
    const int lane = tid & 31;
    const int lm   = lane & 15;           // M (A) / N (B,C,D) within tile
    const int hi   = lane >> 4;           // K-pair selector
    const int c0   = wave * 16;

    const float bval = bias[c0 + lm];
    v8f acc_o, acc_m;
#pragma unroll
    for (int j = 0; j < 8; ++j) { acc_o[j] = bval; acc_m[j] = 0.0f; }

    const float* __restrict__ arow = As + lm * LDS_STRIDE + 2 * hi;
    const float* __restrict__ wr   = Wr + (size_t)(2 * hi) * HDIM + c0 + lm;
    const float* __restrict__ ws   = Ws + (size_t)(2 * hi) * HDIM + c0 + lm;

#pragma unroll 4
    for (int k = 0; k < FDIM; k += 4) {
        // A: 16x4 fp32 fragment from LDS (ds_load_b64, conflict-free)
        v2f a = *(const v2f*)(arow + k);
        // B: 4x16 fp32, rows ka, ka+1 at stride HDIM
        v2f br, bs;
        br.x = wr[(size_t)k * HDIM];
        br.y = wr[(size_t)(k + 1) * HDIM];
        bs.x = ws[(size_t)k * HDIM];
        bs.y = ws[(size_t)(k + 1) * HDIM];
        acc_m = __builtin_amdgcn_wmma_f32_16x16x4_f32(
            false, a, false, br, (short)0, acc_m, false, false);
        acc_o = __builtin_amdgcn_wmma_f32_16x16x4_f32(
            false, a, false, bs, (short)0, acc_o, false, false);
    }

    // C/D layout: VGPR j -> M = j + 8*hi, N = c0 + lm
    float* __restrict__ po = out + (size_t)(r0 + 8 * hi) * HDIM + c0 + lm;
    float* __restrict__ pm = m   + (size_t)(r0 + 8 * hi) * HDIM + c0 + lm;
    if (r0 + 16 <= N) {
#pragma unroll
        for (int j = 0; j < 8; ++j) {
            po[(size_t)j * HDIM] = acc_o[j];
            pm[(size_t)j * HDIM] = acc_m[j];
        }
    } else {
#pragma unroll
        for (int j = 0; j < 8; ++j) {
            if (r0 + 8 * hi + j < N) {
                po[(size_t)j * HDIM] = acc_o[j];
                pm[(size_t)j * HDIM] = acc_m[j];
            }
        }
    }
}

// One wave per edge: out[dst] += m[src] (128 floats, 4 per lane, f32 atomics)
__global__ __launch_bounds__(256) void edge_scatter(
    const int* __restrict__ src, const int* __restrict__ dst,
    const float* __restrict__ m, float* __restrict__ out, int E)
{
    const int e = blockIdx.x * 8 + (threadIdx.x >> 5);
    if (e >= E) return;
    const int lane = threadIdx.x & 31;
    const float4 v = *(const float4*)(m + (size_t)src[e] * HDIM + lane * 4);
    float* o = out + (size_t)dst[e] * HDIM + lane * 4;
    atomicAdd(o + 0, v.x);
    atomicAdd(o + 1, v.y);
    atomicAdd(o + 2, v.z);
    atomicAdd(o + 3, v.w);
}

__global__ __launch_bounds__(256) void zero_buf(float* p, int n)
{
    const int i = blockIdx.x * 256 + threadIdx.x;
    if (i < n) p[i] = 0.0f;
}

// One wave per node: sums[batch[n]] += h[n]; counts[batch[n]] += 1
__global__ __launch_bounds__(256) void pool_scatter(
    const float* __restrict__ h, const int* __restrict__ batch,
    float* __restrict__ sums, float* __restrict__ counts, int N)
{
    const int n = blockIdx.x * 8 + (threadIdx.x >> 5);
    if (n >= N) return;
    const int lane = threadIdx.x & 31;
    const int g = batch[n];
    const float4 v = *(const float4*)(h + (size_t)n * HDIM + lane * 4);
    float* s = sums + (size_t)g * HDIM + lane * 4;
    atomicAdd(s + 0, v.x);
    atomicAdd(s + 1, v.y);
    atomicAdd(s + 2, v.z);
    atomicAdd(s + 3, v.w);
    if (lane == 0) atomicAdd(counts + g, 1.0f);
}

// One block per graph: pooled = sums/count; out = pooled @ W_lin + b_lin
__global__ __launch_bounds__(128) void final_linear(
    const float* __restrict__ sums, const float* __restrict__ counts,
    const float* __restrict__ Wl, const float* __restrict__ bl,
    float* __restrict__ out)
{
    __shared__ float p[HDIM];
    const int g = blockIdx.x;
    const float inv = 1.0f / fmaxf(counts[g], 1.0f);
    p[threadIdx.x] = sums[(size_t)g * HDIM + threadIdx.x] * inv;
    __syncthreads();
    if (threadIdx.x < NCLASS) {
        float acc = bl[threadIdx.x];
#pragma unroll 8
        for (int k = 0; k < HDIM; ++k)
            acc += p[k] * Wl[k * NCLASS + threadIdx.x];
        out[g * NCLASS + threadIdx.x] = acc;
    }
}

extern "C" void kernel_launch(void* const* d_in, const int* in_sizes, int n_in,
                              void* d_out, int out_size, void* d_ws, size_t ws_size,
                              hipStream_t stream)
{
    const float* x      = (const float*)d_in[0];
    const int*   ei     = (const int*)  d_in[1];
    const int*   batch  = (const int*)  d_in[2];
    const float* W_rel  = (const float*)d_in[3];
    const float* W_root = (const float*)d_in[4];
    const float* b      = (const float*)d_in[5];
    const float* W_lin  = (const float*)d_in[6];
    const float* b_lin  = (const float*)d_in[7];
    float* outp = (float*)d_out;

    const int N = in_sizes[0] / FDIM;
    const int E = in_sizes[1] / 2;

    // workspace layout
    float* bufB   = (float*)d_ws;
    float* bufA   = bufB + (size_t)N * HDIM;
    float* bufM   = bufA + (size_t)N * HDIM;
    float* sums   = bufM + (size_t)N * HDIM;
    float* counts = sums + (size_t)NGRAPH * HDIM;

    const int* srcI = ei;
    const int* dstI = ei + E;

    const int gemmBlocks = (N + 15) / 16;
    const int edgeBlocks = (E + 7) / 8;

    const float* cur = x;
    float* pingpong[2] = { bufB, bufA };
    for (int l = 0; l < NLAYER; ++l) {
        float* o = pingpong[l & 1];
        const float* Wr_l = W_rel  + (size_t)l * FDIM * HDIM;
        const float* Ws_l = W_root + (size_t)l * FDIM * HDIM;
        const float* b_l  = b + (size_t)l * HDIM;
        if (l == 0) {
            gemm_dual_wmma<false><<<gemmBlocks, 256, 0, stream>>>(
                cur, Wr_l, Ws_l, b_l, o, bufM, N);
        } else {
            gemm_dual_wmma<true><<<gemmBlocks, 256, 0, stream>>>(
                cur, Wr_l, Ws_l, b_l, o, bufM, N);
        }
        edge_scatter<<<edgeBlocks, 256, 0, stream>>>(srcI, dstI, bufM, o, E);
        cur = o;
    }

    const int poolElems = NGRAPH * HDIM + NGRAPH;
    zero_buf<<<(poolElems + 255) / 256, 256, 0, stream>>>(sums, poolElems);
    pool_scatter<<<(N + 7) / 8, 256, 0, stream>>>(cur, batch, sums, counts, N);
    final_linear<<<NGRAPH, 128, 0, stream>>>(sums, counts, W_lin, b_lin, outp);
}